// OpticalFlow_35158602285814
// MI455X (gfx1250) — compile-verified
//
#include <hip/hip_runtime.h>
#include <cstdint>

// dense_image_warp: image [8,512,512,16] f32 (NHWC), flow [8,512,512,2] f32.
// Memory-bound (≈272 MiB min traffic -> ~12us @ 23.3 TB/s HBM). Strategy:
//   * 4 lanes per pixel, one float4 channel-quad per lane (all VMEM is B128;
//     4 lanes cover one contiguous 64B NHWC pixel row per neighbor).
//   * 64 pixels per 256-thread block; the block's 512B of flow is staged into
//     LDS with the gfx1250 async path (global_load_async_to_lds_b128 +
//     s_wait_asynccnt 0), de-duplicating the 4x-redundant flow reads.
//   * Output is write-once: store with the NT temporal hint so the 128 MiB
//     stream does not evict image lines (gather reuse) from the 192MB L2.
//   * Bilinear blend matches the reference's clamp semantics exactly.

#define NB 8
#define BH 512
#define BW 512
#define BC 16
#define PIX_PER_BLOCK 64   // 256 threads / 4 lanes-per-pixel
#define THREADS 256

typedef float v4f __attribute__((ext_vector_type(4)));

__global__ __launch_bounds__(THREADS)
void OpticalFlow_warp_kernel(const float* __restrict__ image,
                             const float* __restrict__ flow,
                             float* __restrict__ out,
                             long long numPix)
{
    __shared__ float flowBuf[PIX_PER_BLOCK * 2];   // 512 bytes

    const int tid = threadIdx.x;
    const long long pixBase = (long long)blockIdx.x * PIX_PER_BLOCK;

    // ---- Stage this block's flow slab into LDS via the CDNA5 async data path.
    // One wave32: 32 lanes x 16 B = 512 B = 64 pixels * 2 floats.
    if (tid < 32) {
        uint64_t gaddr  = (uint64_t)(uintptr_t)flow
                        + (uint64_t)(pixBase * 2ll * 4ll)
                        + (uint64_t)(tid * 16);
        // Low 32 bits of a flat LDS pointer carry the workgroup-relative
        // LDS byte offset, which is what the async instruction's VDST wants.
        uint32_t ldsOff = (uint32_t)(uintptr_t)(&flowBuf[0]) + (uint32_t)(tid * 16);
        asm volatile("global_load_async_to_lds_b128 %0, %1, off"
                     :: "v"(ldsOff), "v"(gaddr) : "memory");
        asm volatile("s_wait_asynccnt 0" ::: "memory");
    }
    __syncthreads();

    const int pixLocal = tid >> 2;     // 0..63
    const int q        = tid & 3;      // channel quad 0..3
    const long long p  = pixBase + pixLocal;
    if (p >= numPix) return;

    const float dy = flowBuf[pixLocal * 2 + 0];
    const float dx = flowBuf[pixLocal * 2 + 1];

    // Decompose pixel id (H*W = 2^18, W = 2^9).
    const int b   = (int)(p >> 18);
    const int rem = (int)(p & (BH * BW - 1));
    const int y   = rem >> 9;
    const int x   = rem & (BW - 1);

    // Query point = grid - flow (reference: flow[...,0]=dy, flow[...,1]=dx).
    const float qy = (float)y - dy;
    const float qx = (float)x - dx;

    // floor clipped to [0, size-2]; fractional weights clipped to [0,1].
    float fy = floorf(qy); fy = fminf(fmaxf(fy, 0.0f), (float)(BH - 2));
    float fx = floorf(qx); fx = fminf(fmaxf(fx, 0.0f), (float)(BW - 2));
    const float ay = fminf(fmaxf(qy - fy, 0.0f), 1.0f);
    const float ax = fminf(fmaxf(qx - fx, 0.0f), 1.0f);
    const int iy = (int)fy;
    const int ix = (int)fx;

    // Gather the 4 neighbor channel-quads (each a 16B global_load_b128).
    const float* base = image
        + ((((long long)b * BH + iy) * BW + ix) << 4)   // *16 channels
        + (q << 2);                                     // this lane's quad
    const long long rowStride = (long long)BW * BC;     // next y

    const v4f tl = *(const v4f*)(base);
    const v4f tr = *(const v4f*)(base + BC);
    const v4f bl = *(const v4f*)(base + rowStride);
    const v4f br = *(const v4f*)(base + rowStride + BC);

    // top = tl + ax*(tr-tl); bot = bl + ax*(br-bl); out = top + ay*(bot-top)
    v4f res;
    {
        float tx, bx;
        tx = fmaf(ax, tr.x - tl.x, tl.x);
        bx = fmaf(ax, br.x - bl.x, bl.x);
        res.x = fmaf(ay, bx - tx, tx);

        tx = fmaf(ax, tr.y - tl.y, tl.y);
        bx = fmaf(ax, br.y - bl.y, bl.y);
        res.y = fmaf(ay, bx - tx, tx);

        tx = fmaf(ax, tr.z - tl.z, tl.z);
        bx = fmaf(ax, br.z - bl.z, bl.z);
        res.z = fmaf(ay, bx - tx, tx);

        tx = fmaf(ax, tr.w - tl.w, tl.w);
        bx = fmaf(ax, br.w - bl.w, bl.w);
        res.w = fmaf(ay, bx - tx, tx);
    }

    // Coalesced write-once stream: non-temporal B128 store (don't pollute L2's
    // image working set; 4 lanes cover the pixel's 64B channel row).
    v4f* dst = (v4f*)(out + (p << 4) + (q << 2));
    __builtin_nontemporal_store(res, dst);
}

extern "C" void kernel_launch(void* const* d_in, const int* in_sizes, int n_in,
                              void* d_out, int out_size, void* d_ws, size_t ws_size,
                              hipStream_t stream) {
    const float* image = (const float*)d_in[0];  // [8,512,512,16]
    const float* flow  = (const float*)d_in[1];  // [8,512,512,2]
    float* out = (float*)d_out;                  // [8,512,512,16]

    const long long numPix = (long long)NB * BH * BW;        // 2,097,152
    const unsigned blocks  = (unsigned)((numPix + PIX_PER_BLOCK - 1) / PIX_PER_BLOCK);

    OpticalFlow_warp_kernel<<<blocks, THREADS, 0, stream>>>(image, flow, out, numPix);
}